// BayesianLinear_73237782332008
// MI455X (gfx1250) — compile-verified
//
#include <hip/hip_runtime.h>
#include <hip/hip_bf16.h>
#include <stdint.h>

constexpr int S_   = 32;
constexpr int IN_  = 2048;
constexpr int OUT_ = 2048;

typedef float v2f __attribute__((ext_vector_type(2)));
typedef float v8f __attribute__((ext_vector_type(8)));

// ---------------------------------------------------------------------------
// Kernel A: out[s,o] = sum_i x[s,i] * mu[o,i]   via V_WMMA_F32_16X16X4_F32
// One wave per 16x16 output tile. A = x tile (M=s rows), B = mu^T tile (N=o cols).
// ---------------------------------------------------------------------------
__global__ __launch_bounds__(32) void bayes_gemm_mu_wmma(
    const float* __restrict__ x,      // (S, IN)
    const float* __restrict__ mu,     // (OUT, IN)
    float* __restrict__ out)          // (S, OUT)
{
    const int o0   = blockIdx.x * 16;
    const int s0   = blockIdx.y * 16;
    const int lane = threadIdx.x;     // 0..31, wave32
    const int hl   = lane >> 4;       // lane half: selects K pair {0,1} vs {2,3}
    const int idx  = lane & 15;       // M index for A, N index for B

    // A lane layout (16x4 f32): v0,v1 = A[M=idx, K=2*hl+0 / +1]
    const float* xrow  = x  + (size_t)(s0 + idx) * IN_ + 2 * hl;
    // B lane layout (4x16 f32): v0,v1 = B[K=2*hl+0 / +1, N=idx] = mu[o0+idx, k+K]
    const float* murow = mu + (size_t)(o0 + idx) * IN_ + 2 * hl;

    v8f acc = {};
    #pragma unroll 4
    for (int k = 0; k < IN_; k += 4) {
        v2f a = *(const v2f*)(xrow + k);
        v2f b = *(const v2f*)(murow + k);
        acc = __builtin_amdgcn_wmma_f32_16x16x4_f32(
            /*neg_a=*/false, a, /*neg_b=*/false, b,
            /*c_mod=*/(short)0, acc, /*reuse_a=*/false, /*reuse_b=*/false);
    }

    // D layout: element (M = v + 8*hl, N = idx)
    #pragma unroll
    for (int v = 0; v < 8; ++v) {
        out[(size_t)(s0 + v + 8 * hl) * OUT_ + (o0 + idx)] = acc[v];
    }
}

// ---------------------------------------------------------------------------
// Philox4x32-10 counter-based RNG + Box-Muller (in-register Gaussian draws)
// ---------------------------------------------------------------------------
__device__ __forceinline__ uint4 philox4x32(uint4 c, uint32_t k0, uint32_t k1) {
    const uint32_t M0 = 0xD2511F53u, M1 = 0xCD9E8D57u;
    const uint32_t W0 = 0x9E3779B9u, W1 = 0xBB67AE85u;
    #pragma unroll
    for (int r = 0; r < 10; ++r) {
        uint32_t lo0 = M0 * c.x, hi0 = __umulhi(M0, c.x);
        uint32_t lo1 = M1 * c.z, hi1 = __umulhi(M1, c.z);
        uint4 n;
        n.x = hi1 ^ c.y ^ k0;
        n.y = lo1;
        n.z = hi0 ^ c.w ^ k1;
        n.w = lo0;
        c = n;
        k0 += W0; k1 += W1;
    }
    return c;
}

__device__ __forceinline__ float u01(uint32_t u) {
    // (0,1): 24-bit mantissa draw, offset by half-ulp so log() is safe
    return (float)(u >> 8) * 0x1.0p-24f + 0x1.0p-25f;
}

__device__ __forceinline__ float4 gauss4(uint4 r) {
    const float TWO_PI = 6.28318530717958647692f;
    float r1 = __builtin_sqrtf(-2.0f * __logf(u01(r.x)));
    float r2 = __builtin_sqrtf(-2.0f * __logf(u01(r.z)));
    float s1, c1, s2, c2;
    __sincosf(TWO_PI * u01(r.y), &s1, &c1);
    __sincosf(TWO_PI * u01(r.w), &s2, &c2);
    return make_float4(r1 * c1, r1 * s1, r2 * c2, r2 * s2);
}

// ---------------------------------------------------------------------------
// Kernel B: out[s,o] += sum_i sigma[o,i]*eps(s,o,i)*x[s,i]
//                     + bias_mu[o] + bias_sigma[o]*eps_b(s,o)
// Block = 8 waves, one sample s; x[s,:] staged in LDS; each wave owns an o,
// lanes cover 4 consecutive i's (one Philox call -> 4 normals) per step.
// ---------------------------------------------------------------------------
__global__ __launch_bounds__(256) void bayes_noise_bias(
    const float* __restrict__ x,          // (S, IN)
    const float* __restrict__ sigma,      // (OUT, IN)
    const float* __restrict__ bias_mu,    // (OUT)
    const float* __restrict__ bias_sigma, // (OUT)
    float* __restrict__ out)              // (S, OUT)
{
    __shared__ float xs[IN_];             // 8 KB of 320 KB LDS

    const int s = blockIdx.y;
    for (int i = threadIdx.x; i < IN_; i += 256) xs[i] = x[(size_t)s * IN_ + i];
    __syncthreads();

    const int wave   = threadIdx.x >> 5;  // 0..7
    const int lane   = threadIdx.x & 31;
    const int o_base = blockIdx.x * 32;   // 32 outputs per block, 4 per wave

    #pragma unroll
    for (int ow = 0; ow < 4; ++ow) {
        const int o = o_base + ow * 8 + wave;
        const float* srow = sigma + (size_t)o * IN_;

        float acc = 0.0f;
        for (int it = 0; it < IN_ / 128; ++it) {   // 128 i's per wave-step
            const int i = it * 128 + lane * 4;
            float4 sg = *(const float4*)(srow + i);        // coalesced b128
            float4 xv = *(const float4*)(&xs[i]);          // LDS b128
            uint4  rn = philox4x32(make_uint4((uint32_t)(i >> 2), (uint32_t)o,
                                              (uint32_t)s, 0x0000002Au),
                                   0xC0FFEE42u, 0xDEADBEEFu);
            float4 e  = gauss4(rn);
            acc += sg.x * e.x * xv.x;
            acc += sg.y * e.y * xv.y;
            acc += sg.z * e.z * xv.z;
            acc += sg.w * e.w * xv.w;
        }

        // wave32 butterfly reduction
        #pragma unroll
        for (int off = 16; off > 0; off >>= 1)
            acc += __shfl_xor(acc, off, 32);

        if (lane == 0) {
            uint4  rb = philox4x32(make_uint4((uint32_t)o, (uint32_t)s,
                                              0x000B1A5Eu, 0x0000002Bu),
                                   0xC0FFEE42u, 0xDEADBEEFu);
            float4 eb = gauss4(rb);
            const size_t oi = (size_t)s * OUT_ + o;
            out[oi] += acc + bias_mu[o] + bias_sigma[o] * eb.x;
        }
    }
}

// ---------------------------------------------------------------------------
extern "C" void kernel_launch(void* const* d_in, const int* in_sizes, int n_in,
                              void* d_out, int out_size, void* d_ws, size_t ws_size,
                              hipStream_t stream) {
    const float* x          = (const float*)d_in[0];
    const float* weight_mu  = (const float*)d_in[1];
    const float* weight_sig = (const float*)d_in[2];
    const float* bias_mu    = (const float*)d_in[3];
    const float* bias_sig   = (const float*)d_in[4];
    float*       out        = (float*)d_out;

    dim3 gA(OUT_ / 16, S_ / 16, 1);           // 128 x 2 tiles, 1 wave each
    bayes_gemm_mu_wmma<<<gA, dim3(32, 1, 1), 0, stream>>>(x, weight_mu, out);

    dim3 gB(OUT_ / 32, S_, 1);                // 64 x 32 blocks, 8 waves each
    bayes_noise_bias<<<gB, dim3(256, 1, 1), 0, stream>>>(
        x, weight_sig, bias_mu, bias_sig, out);
}